// AUCLoss_1159641170386
// MI455X (gfx1250) — compile-verified
//
#include <hip/hip_runtime.h>

#define BATCH 64
#define CLS   206
#define N_    (BATCH * CLS)   // 13184
#define TPB   256
#define JSPLIT 8
#define GX    ((N_ + TPB - 1) / TPB)   // 52

typedef __attribute__((ext_vector_type(2))) float v2f;
typedef __attribute__((ext_vector_type(8))) float v8f;

// stable softplus(z) = max(z,0) + log(1 + exp(-|z|)), native trans ops
__device__ __forceinline__ float softplus_f(float z) {
    return fmaxf(z, 0.f) + __logf(1.f + __expf(-fabsf(z)));
}

// Each block: 256 rows (i) x one j-chunk. Writes one deterministic partial
// sum per block into ws[blockIdx.y * GX + blockIdx.x].
__global__ __launch_bounds__(TPB) void auc_pairs_kernel(
    const float* __restrict__ preds,
    const float* __restrict__ sw,
    const int*   __restrict__ labels,
    float*       __restrict__ partials)
{
    __shared__ float s_p[TPB];
    __shared__ float s_nw[TPB];
    __shared__ float s_wave[TPB / 32];

    const int tid = threadIdx.x;
    const int i   = blockIdx.x * TPB + tid;

    float p_i = 0.f, pw_i = 0.f;
    if (i < N_) {
        p_i  = preds[i];
        pw_i = (labels[i] == 1) ? sw[i / CLS] : 0.f;
    }

    const int jchunk = (N_ + JSPLIT - 1) / JSPLIT;   // 1648
    const int j0 = blockIdx.y * jchunk;
    const int j1 = min(j0 + jchunk, N_);

    float sum = 0.f;
    for (int jb = j0; jb < j1; jb += TPB) {
        const int j = jb + tid;
        float pj = 0.f, nwj = 0.f;
        if (j < j1) {
            pj  = preds[j];
            nwj = (labels[j] == 0) ? sw[j / CLS] : 0.f;  // 0-weight pads tile
        }
        if (jb + TPB < j1) {
            __builtin_prefetch(&preds[jb + TPB + tid], 0, 0);   // global_prefetch_b8
            __builtin_prefetch(&labels[jb + TPB + tid], 0, 0);
        }
        __syncthreads();
        s_p[tid]  = pj;
        s_nw[tid] = nwj;
        __syncthreads();

        if (pw_i != 0.f) {            // whole row masked out -> skip compute
            #pragma unroll 8
            for (int k = 0; k < TPB; ++k) {
                const float z = s_p[k] - p_i;               // -(p_i - p_j)
                sum = fmaf(softplus_f(z), s_nw[k], sum);
            }
        }
    }

    const float tot = pw_i * sum;     // hoisted row weight

    // ---- wave32 reduction via WMMA: D = A(16x4) * ones(4x16) + 0 ----
    // A layout (32-bit 16x4): lanes 0-15 -> K=0 (VGPR0), K=1 (VGPR1);
    // lanes 16-31 -> K=2,3.  Put each lane's partial in VGPR0, zero VGPR1.
    // All-ones B is layout-invariant.  D[m][n] = part[m] + part[m+16].
    v2f a; a[0] = tot; a[1] = 0.f;
    v2f b; b[0] = 1.f; b[1] = 1.f;
    v8f c = {};
    v8f d = __builtin_amdgcn_wmma_f32_16x16x4_f32(
        /*neg_a=*/false, a, /*neg_b=*/false, b,
        /*c_mod=*/(short)0, c, /*reuse_a=*/false, /*reuse_b=*/false);

    // lanes 0-15 hold rows 0-7 of their column, lanes 16-31 hold rows 8-15
    float t = d[0] + d[1] + d[2] + d[3] + d[4] + d[5] + d[6] + d[7];
    t += __shfl_xor(t, 16, 32);       // now full 32-lane sum on every lane

    if ((tid & 31) == 0) s_wave[tid >> 5] = t;
    __syncthreads();
    if (tid == 0) {
        float bs = 0.f;
        #pragma unroll
        for (int wv = 0; wv < TPB / 32; ++wv) bs += s_wave[wv];  // fixed order
        partials[blockIdx.y * GX + blockIdx.x] = bs;
    }
}

// Deterministic finalize: fixed-order sum of 416 block partials + pos/neg
// counts, then the normalization divide.
__global__ __launch_bounds__(TPB) void auc_finalize_kernel(
    const int*   __restrict__ labels,
    const float* __restrict__ partials,
    float*       __restrict__ out)
{
    __shared__ int s_pos, s_neg;
    if (threadIdx.x == 0) { s_pos = 0; s_neg = 0; }
    __syncthreads();

    int lp = 0, ln = 0;
    for (int idx = threadIdx.x; idx < N_; idx += TPB) {
        const int lab = labels[idx];
        lp += (lab == 1);
        ln += (lab == 0);
    }
    atomicAdd(&s_pos, lp);   // integer LDS atomics: order-independent
    atomicAdd(&s_neg, ln);
    __syncthreads();

    if (threadIdx.x == 0) {
        float total = 0.f;
        for (int k = 0; k < GX * JSPLIT; ++k) total += partials[k];  // fixed order
        out[0] = total / ((float)s_pos * (float)s_neg);
    }
}

extern "C" void kernel_launch(void* const* d_in, const int* in_sizes, int n_in,
                              void* d_out, int out_size, void* d_ws, size_t ws_size,
                              hipStream_t stream)
{
    const float* preds  = (const float*)d_in[0];   // [64,206] f32
    const float* sw     = (const float*)d_in[1];   // [64] f32
    const int*   labels = (const int*)d_in[2];     // [64,206] i32
    float* out      = (float*)d_out;               // scalar f32
    float* partials = (float*)d_ws;                // GX*JSPLIT floats (1.7 KB)

    dim3 grid(GX, JSPLIT);
    auc_pairs_kernel<<<grid, TPB, 0, stream>>>(preds, sw, labels, partials);
    auc_finalize_kernel<<<1, TPB, 0, stream>>>(labels, partials, out);
}